// V5ArithmeticMoE_62380105007632
// MI455X (gfx1250) — compile-verified
//
#include <hip/hip_runtime.h>
#include <hip/hip_bf16.h>

// ---------------------------------------------------------------------------
// Fused streaming kernel for the arithmetic-MoE:
//
//   x (32*2048 tokens x 512 f32)  --async DMA-->  LDS  --patch-->  out
//
// Each block moves 1024 float4 (16 KB) = exactly 8 complete tokens:
//   1) 4x global_load_async_to_lds_b128 per thread (coalesced 16B/lane)
//   2) s_wait_asynccnt 0 + barrier: block's 8 tokens contiguous in LDS
//   3) lane 0 of each of the 8 waves patches its token's result slots in LDS
//      (reproducing the reference float semantics for the one-hot-selected op)
//   4) barrier, then 4x global_store_async_from_lds_b128 per thread
//
// The bulk 268 MB of traffic stays on the CDNA5 async DMA path (ASYNCcnt),
// and the tiny fix-up rides along in LDS instead of a second kernel.
// ---------------------------------------------------------------------------
__global__ __launch_bounds__(256) void moe_fused_async(const float4* __restrict__ in,
                                                       float4* __restrict__ out) {
    __shared__ float4 buf[1024];            // 16 KB staging, LDS offset 0
    float* __restrict__ buff = (float*)buf; // scalar view for the fix-up

    const unsigned t = threadIdx.x;
    const size_t base = (size_t)blockIdx.x * 1024u + t;

    const float4* s0 = in + base;
    const float4* s1 = s0 + 256;
    const float4* s2 = s0 + 512;
    const float4* s3 = s0 + 768;
    float4* d0 = out + base;
    float4* d1 = d0 + 256;
    float4* d2 = d0 + 512;
    float4* d3 = d0 + 768;

    const unsigned l0 = t * 16u;            // LDS byte offset of slot 0
    const unsigned l1 = l0 + 4096u;
    const unsigned l2 = l0 + 8192u;
    const unsigned l3 = l0 + 12288u;

    // ---- stage in: global -> LDS (async DMA, per-lane addresses) ----------
    asm volatile(
        "global_load_async_to_lds_b128 %0, %4, off\n\t"
        "global_load_async_to_lds_b128 %1, %5, off\n\t"
        "global_load_async_to_lds_b128 %2, %6, off\n\t"
        "global_load_async_to_lds_b128 %3, %7, off\n\t"
        "s_wait_asynccnt 0x0\n\t"
        :
        : "v"(l0), "v"(l1), "v"(l2), "v"(l3),
          "v"(s0), "v"(s1), "v"(s2), "v"(s3)
        : "memory");
    __syncthreads();   // all waves' async-loaded LDS data now visible

    // ---- per-token fix-up in LDS: wave w (lane 0) patches token w ---------
    if ((t & 31u) == 0u) {
        const unsigned tokbase = (t >> 5) * 512u;   // token index within block
        float a[8], b[8];
#pragma unroll
        for (int i = 0; i < 8; ++i) {
            a[i] = buff[tokbase + i];
            b[i] = buff[tokbase + 8 + i];
        }
        // Gate slots (OPC = 104): or/xor/and = 118/119/120, eq..ge = 121..126,
        // add/sub = 129/130, div/mod = 132/133.
        const float g_or  = buff[tokbase + 118], g_xor = buff[tokbase + 119];
        const float g_and = buff[tokbase + 120];
        const float g_eq  = buff[tokbase + 121], g_ne  = buff[tokbase + 122];
        const float g_lt  = buff[tokbase + 123], g_gt  = buff[tokbase + 124];
        const float g_le  = buff[tokbase + 125], g_ge  = buff[tokbase + 126];
        const float g_add = buff[tokbase + 129], g_sub = buff[tokbase + 130];
        const float g_div = buff[tokbase + 132], g_mod = buff[tokbase + 133];

        if (g_add + g_sub > 0.5f) {
            // Nibble-serial add (carry) and sub (borrow), gated blend.
            float carry = 0.0f, borrow = 0.0f;
#pragma unroll
            for (int i = 0; i < 8; ++i) {
                float s  = a[i] + b[i] + carry;
                float c  = (s >= 16.0f) ? 1.0f : 0.0f;
                float ad = s - 16.0f * c;
                carry = c;
                float d  = a[i] - b[i] - borrow;
                float nb = (d < 0.0f) ? 1.0f : 0.0f;
                float sb = d + 16.0f * nb;
                borrow = nb;
                buff[tokbase + 16 + i] = g_add * ad + g_sub * sb;
            }
        } else if (g_or + g_xor + g_and > 0.5f) {
            // Per-nibble bit extraction via fmod, as in the reference.
#pragma unroll
            for (int i = 0; i < 8; ++i) {
                float andv = 0.0f, orv = 0.0f, xorv = 0.0f;
                float p = 1.0f;
#pragma unroll
                for (int j = 0; j < 4; ++j) {
                    float abit = (fmodf(a[i], 2.0f * p) >= p) ? 1.0f : 0.0f;
                    float bbit = (fmodf(b[i], 2.0f * p) >= p) ? 1.0f : 0.0f;
                    float both = abit * bbit;
                    andv += both * p;
                    orv  += (abit + bbit - both) * p;
                    xorv += (abit + bbit - 2.0f * both) * p;
                    p *= 2.0f;
                }
                buff[tokbase + 16 + i] = g_and * andv + g_or * orv + g_xor * xorv;
            }
        } else {
            // Value-based ops: weighted sums with w = 16^i.
            float av = 0.0f, bv = 0.0f, w = 1.0f;
#pragma unroll
            for (int i = 0; i < 8; ++i) {
                av += a[i] * w;
                bv += b[i] * w;
                w *= 16.0f;
            }
            if (g_div + g_mod > 0.5f) {
                float bc = fmaxf(bv, 1.0f);
                float q  = floorf(av / bc);
                float r  = av - q * bc;
                float w2 = 1.0f;
#pragma unroll
                for (int i = 0; i < 8; ++i) {
                    float dv = fmodf(floorf(q / w2), 16.0f);
                    float mv = fmodf(floorf(r / w2), 16.0f);
                    buff[tokbase + 16 + i] = g_div * dv + g_mod * mv;
                    w2 *= 16.0f;
                }
            } else {
                // Comparison: flags to 152..157 as (eq,lt,gt,ne,le,ge);
                // res0 to slot 16 weighted in (eq,ne,lt,gt,le,ge) order.
                float eq = (fabsf(av - bv) < 0.5f) ? 1.0f : 0.0f;
                float lt = (av < bv) ? 1.0f : 0.0f;
                float gt = (av > bv) ? 1.0f : 0.0f;
                float ne = 1.0f - eq;
                float le = eq + lt - eq * lt;
                float ge = eq + gt - eq * gt;
                buff[tokbase + 152] = eq;
                buff[tokbase + 153] = lt;
                buff[tokbase + 154] = gt;
                buff[tokbase + 155] = ne;
                buff[tokbase + 156] = le;
                buff[tokbase + 157] = ge;
                buff[tokbase + 16] = g_eq * eq + g_ne * ne + g_lt * lt +
                                     g_gt * gt + g_le * le + g_ge * ge;
            }
        }
    }
    __syncthreads();   // fix-up DS writes drained (s_wait_dscnt) + visible

    // ---- drain out: LDS -> global (async DMA) -----------------------------
    asm volatile(
        "global_store_async_from_lds_b128 %4, %0, off\n\t"
        "global_store_async_from_lds_b128 %5, %1, off\n\t"
        "global_store_async_from_lds_b128 %6, %2, off\n\t"
        "global_store_async_from_lds_b128 %7, %3, off\n\t"
        :
        : "v"(l0), "v"(l1), "v"(l2), "v"(l3),
          "v"(d0), "v"(d1), "v"(d2), "v"(d3)
        : "memory");
    // S_ENDPGM performs an implicit wait-idle, draining the async stores.
}

extern "C" void kernel_launch(void* const* d_in, const int* in_sizes, int n_in,
                              void* d_out, int out_size, void* d_ws, size_t ws_size,
                              hipStream_t stream) {
    (void)in_sizes; (void)n_in; (void)d_ws; (void)ws_size;
    const float* x = (const float*)d_in[0];
    float* out = (float*)d_out;

    // out_size = 32*2048*512 = 33,554,432 floats = 8,388,608 float4.
    // 1024 float4 (= 8 whole tokens) per block -> 8192 blocks.
    const int n4 = out_size / 4;
    const int blocks = n4 / 1024;
    moe_fused_async<<<blocks, 256, 0, stream>>>((const float4*)x, (float4*)out);
}